// GCN_Leo_9448928051730
// MI455X (gfx1250) — compile-verified
//
#include <hip/hip_runtime.h>
#include <hip/hip_bf16.h>

typedef float v2f __attribute__((ext_vector_type(2)));
typedef float v8f __attribute__((ext_vector_type(8)));

#define DEV_ATOMIC_ADD(p, v) \
    __hip_atomic_fetch_add((p), (v), __ATOMIC_RELAXED, __HIP_MEMORY_SCOPE_AGENT)

// ---------------------------------------------------------------------------
// Utility: zero-fill
// ---------------------------------------------------------------------------
__global__ void gcn_zero_f32(float* __restrict__ p, long long n) {
    long long i = (long long)blockIdx.x * blockDim.x + threadIdx.x;
    if (i < n) p[i] = 0.0f;
}

// ---------------------------------------------------------------------------
// Degree accumulation: one thread per edge
// ---------------------------------------------------------------------------
__global__ void gcn_degrees(const int* __restrict__ src, const int* __restrict__ dst,
                            float* __restrict__ deg_out, float* __restrict__ deg_in,
                            int nEdges) {
    int e = blockIdx.x * blockDim.x + threadIdx.x;
    if (e < nEdges) {
        DEV_ATOMIC_ADD(&deg_out[src[e]], 1.0f);
        DEV_ATOMIC_ADD(&deg_in[dst[e]], 1.0f);
    }
}

// deg -> clip(deg,1)^-0.5   (in place, covers both deg arrays back to back)
__global__ void gcn_norms(float* __restrict__ p, long long n) {
    long long i = (long long)blockIdx.x * blockDim.x + threadIdx.x;
    if (i < n) p[i] = rsqrtf(fmaxf(p[i], 1.0f));
}

// ---------------------------------------------------------------------------
// WMMA fp32 GEMM:  Y[nRows, NCOLS] = (X * rowScale?) @ W   (W is K x NCOLS)
//   - uses V_WMMA_F32_16X16X4_F32 (CDNA5 fp32 matrix pipe)
//   - 128 threads = 4 waves; wave w owns rows [blk*64 + w*16, +16)
//   - W staged in LDS, zero-padded to NTILES*16 columns
//   - each wave keeps NTILES accumulator tiles so each A-frag is loaded once
//   - stores: wave-uniform row-bound branch; full blocks take an
//     unpredicated path of 8 immediate-offset global_store_b32 per tile
// ---------------------------------------------------------------------------
template <int K, int NCOLS, int NTILES, bool SCALE>
__global__ __launch_bounds__(128) void gcn_gemm_wmma(
    const float* __restrict__ X,        // [nRows, K]
    const float* __restrict__ W,        // [K, NCOLS], row-major
    const float* __restrict__ rowScale, // [nRows] (ignored if !SCALE)
    float* __restrict__ Y,              // [nRows, NCOLS]
    int nRows) {
    constexpr int NPAD  = NTILES * 16;
    constexpr bool FULLC = (NCOLS == NPAD);  // column predicate needed?
    __shared__ float sW[K][NPAD];

    // Cooperative load of W into LDS (zero-pad columns >= NCOLS)
    for (int i = threadIdx.x; i < K * NPAD; i += 128) {
        int k = i / NPAD;
        int n = i % NPAD;
        sW[k][n] = (n < NCOLS) ? W[k * NCOLS + n] : 0.0f;
    }
    __syncthreads();

    const int wave = threadIdx.x >> 5;
    const int lane = threadIdx.x & 31;
    const int lrow = lane & 15;
    const int hi   = lane >> 4;              // 0: K,K+1   1: K+2,K+3
    const int m0   = blockIdx.x * 64 + wave * 16;

    // Clamp row for loads so EXEC stays all-ones around the WMMAs.
    int row  = m0 + lrow;
    int rowc = row < nRows ? row : nRows - 1;
    const float sc = SCALE ? rowScale[rowc] : 1.0f;
    const float* __restrict__ xrow = X + (long long)rowc * K;

    v8f acc[NTILES] = {};

    for (int k = 0; k < K; k += 4) {
        const int koff = k + hi * 2;
        v2f a;
        a.x = xrow[koff] * sc;
        a.y = xrow[koff + 1] * sc;
#pragma unroll
        for (int t = 0; t < NTILES; ++t) {
            v2f b;
            b.x = sW[koff][t * 16 + lrow];
            b.y = sW[koff + 1][t * 16 + lrow];
            acc[t] = __builtin_amdgcn_wmma_f32_16x16x4_f32(
                false, a, false, b, (short)0, acc[t], false, false);
        }
    }

    // C/D layout: element i of tile t -> row m0 + i + hi*8, col t*16 + lrow
    const int r0 = m0 + hi * 8;
    if (m0 + 16 <= nRows) {            // wave-uniform fast path (scalar branch)
#pragma unroll
        for (int t = 0; t < NTILES; ++t) {
            const int cc = t * 16 + lrow;
            float* __restrict__ yp = Y + (long long)r0 * NCOLS + cc;
            if (FULLC || cc < NCOLS) { // single EXEC region around 8 stores
#pragma unroll
                for (int i = 0; i < 8; ++i) yp[i * NCOLS] = acc[t][i];
            }
        }
    } else {                            // tail block: per-element predication
#pragma unroll
        for (int t = 0; t < NTILES; ++t) {
            const int cc = t * 16 + lrow;
            float* __restrict__ yp = Y + (long long)r0 * NCOLS + cc;
#pragma unroll
            for (int i = 0; i < 8; ++i) {
                if (r0 + i < nRows && (FULLC || cc < NCOLS))
                    yp[i * NCOLS] = acc[t][i];
            }
        }
    }
}

// ---------------------------------------------------------------------------
// Edge scatter-add: out[dst[e], :] += H[src[e], :]
// One thread per (edge, 4-channel group): float4 gather + 4 f32 atomics.
// ---------------------------------------------------------------------------
template <int C>
__global__ void gcn_scatter_add(const float* __restrict__ H,
                                const int* __restrict__ src,
                                const int* __restrict__ dst,
                                float* __restrict__ out, int nEdges) {
    constexpr int G = C / 4;
    long long idx = (long long)blockIdx.x * blockDim.x + threadIdx.x;
    int e = (int)(idx / G);
    int g = (int)(idx % G) * 4;
    if (e < nEdges) {
        int s = src[e];
        int d = dst[e];
        const float4 v = *(const float4*)(H + (long long)s * C + g);
        float* o = out + (long long)d * C + g;
        DEV_ATOMIC_ADD(o + 0, v.x);
        DEV_ATOMIC_ADD(o + 1, v.y);
        DEV_ATOMIC_ADD(o + 2, v.z);
        DEV_ATOMIC_ADD(o + 3, v.w);
    }
}

// ---------------------------------------------------------------------------
// X2 = relu(agg * norm_in + b1) * norm_out      (fused: pre-applies next
// layer's source normalization)                  C = 64
// ---------------------------------------------------------------------------
__global__ void gcn_finalize1(const float* __restrict__ agg,
                              const float* __restrict__ norm_in,
                              const float* __restrict__ norm_out,
                              const float* __restrict__ b1,
                              float* __restrict__ X2, long long total) {
    long long i = (long long)blockIdx.x * blockDim.x + threadIdx.x;
    if (i < total) {
        int n = (int)(i >> 6);
        int c = (int)(i & 63);
        float h = agg[i] * norm_in[n] + b1[c];
        h = fmaxf(h, 0.0f);
        X2[i] = h * norm_out[n];
    }
}

// out = out * norm_in + b2   (in place)   C = 40
__global__ void gcn_finalize2(float* __restrict__ out,
                              const float* __restrict__ norm_in,
                              const float* __restrict__ b2, long long total) {
    long long i = (long long)blockIdx.x * blockDim.x + threadIdx.x;
    if (i < total) {
        int n = (int)(i / 40);
        int c = (int)(i % 40);
        out[i] = out[i] * norm_in[n] + b2[c];
    }
}

// ---------------------------------------------------------------------------
// Host launcher
// Inputs: 0 features[N,128] f32, 1 src[E] i32, 2 dst[E] i32,
//         3 W1[128,64] f32, 4 b1[64] f32, 5 W2[64,40] f32, 6 b2[40] f32
// Output: [N, 40] f32
// ---------------------------------------------------------------------------
extern "C" void kernel_launch(void* const* d_in, const int* in_sizes, int n_in,
                              void* d_out, int out_size, void* d_ws, size_t ws_size,
                              hipStream_t stream) {
    constexpr int IN = 128, HID = 64, OUT = 40;
    const float* features = (const float*)d_in[0];
    const int*   src      = (const int*)d_in[1];
    const int*   dst      = (const int*)d_in[2];
    const float* W1       = (const float*)d_in[3];
    const float* b1       = (const float*)d_in[4];
    const float* W2       = (const float*)d_in[5];
    const float* b2       = (const float*)d_in[6];
    float*       out      = (float*)d_out;

    const int nNodes = in_sizes[0] / IN;
    const int nEdges = in_sizes[1];

    // Workspace layout (floats)
    float* w        = (float*)d_ws;
    float* norm_out = w;                                  // nNodes (deg_out -> norm)
    float* norm_in  = w + nNodes;                         // nNodes (deg_in  -> norm)
    float* bufA     = w + 2LL * nNodes;                   // nNodes*HID  (H1, then X2)
    float* bufB     = bufA + (long long)nNodes * HID;     // nNodes*HID  (agg1)
    float* bufC     = bufB + (long long)nNodes * HID;     // nNodes*OUT  (H2)

    const int T = 256;
    auto blocks = [](long long n, int t) { return (int)((n + t - 1) / t); };

    // 1) zero degree buffers, agg1 buffer, output (used as agg2)
    gcn_zero_f32<<<blocks(2LL * nNodes, T), T, 0, stream>>>(w, 2LL * nNodes);
    gcn_zero_f32<<<blocks((long long)nNodes * HID, T), T, 0, stream>>>(
        bufB, (long long)nNodes * HID);
    gcn_zero_f32<<<blocks((long long)nNodes * OUT, T), T, 0, stream>>>(
        out, (long long)nNodes * OUT);

    // 2) degrees + norms
    gcn_degrees<<<blocks(nEdges, T), T, 0, stream>>>(src, dst, norm_out, norm_in, nEdges);
    gcn_norms<<<blocks(2LL * nNodes, T), T, 0, stream>>>(w, 2LL * nNodes);

    // 3) H1 = (X * norm_out) @ W1   — fp32 WMMA
    const int gemmBlocks = (nNodes + 63) / 64;
    gcn_gemm_wmma<IN, HID, HID / 16, true><<<gemmBlocks, 128, 0, stream>>>(
        features, W1, norm_out, bufA, nNodes);

    // 4) agg1[dst] += H1[src]
    gcn_scatter_add<HID><<<blocks((long long)nEdges * (HID / 4), T), T, 0, stream>>>(
        bufA, src, dst, bufB, nEdges);

    // 5) X2 = relu(agg1 * norm_in + b1) * norm_out
    gcn_finalize1<<<blocks((long long)nNodes * HID, T), T, 0, stream>>>(
        bufB, norm_in, norm_out, b1, bufA, (long long)nNodes * HID);

    // 6) H2 = X2 @ W2   (40 cols padded to 48 in LDS) — fp32 WMMA
    gcn_gemm_wmma<HID, OUT, 3, false><<<gemmBlocks, 128, 0, stream>>>(
        bufA, W2, nullptr, bufC, nNodes);

    // 7) out[dst] += H2[src]
    gcn_scatter_add<OUT><<<blocks((long long)nEdges * (OUT / 4), T), T, 0, stream>>>(
        bufC, src, dst, out, nEdges);

    // 8) out = out * norm_in + b2
    gcn_finalize2<<<blocks((long long)nNodes * OUT, T), T, 0, stream>>>(
        out, norm_in, b2, (long long)nNodes * OUT);
}